// Model_33921651704475
// MI455X (gfx1250) — compile-verified
//
#include <hip/hip_runtime.h>

typedef __attribute__((ext_vector_type(2))) float v2f;
typedef __attribute__((ext_vector_type(4))) float v4f;
typedef __attribute__((ext_vector_type(8))) float v8f;

#define CIN  32
#define COUT 16
#define HIN  256
#define WIN  256
#define HP   128
#define WP   128
#define OUT_PAD 132   // padded row stride for the output-transpose LDS tile

// One block handles one (n, hp) pooled row: reads 2 input rows x 32 channels,
// produces out[n, 0..15, hp, 0..127].
__global__ __launch_bounds__(256) void fused_bn_relu_pool_conv1x1(
    const float* __restrict__ x,        // [32][32][256][256]
    const float* __restrict__ w,        // [32][16]  (C_in, C_out)
    const float* __restrict__ bn_scale, // [32]
    const float* __restrict__ bn_bias,  // [32]
    float* __restrict__ out)            // [32][16][128][128]
{
    __shared__ float lds_pool[CIN * WP];       // 16 KB: pooled[c][wp]
    __shared__ float lds_out[COUT * OUT_PAD];  // 8.25 KB: outtile[d][wp] (padded)

    const int t   = threadIdx.x;
    const int bid = blockIdx.x;
    const int n   = bid >> 7;    // / 128
    const int hp  = bid & 127;
    const int row = hp * 2;

    // ---------- Stage 1: BN(affine) + ReLU + 2x2 avg-pool -> LDS ----------
    // idx enumerates (c, wp-pair); consecutive lanes -> consecutive 16B global
    // chunks (global_load_b128), fully coalesced. x is streamed once (268 MB >
    // 192 MB L2), so use non-temporal loads to avoid rinsing L2.
    #pragma unroll
    for (int i = 0; i < 8; ++i) {
        int idx = i * 256 + t;           // 0..2047
        int c   = idx >> 6;              // 0..31
        int wq  = idx & 63;              // pooled-column pair: wp = 2*wq, 2*wq+1
        const v4f* p0 = (const v4f*)(x + (((size_t)(n * CIN + c) * HIN + row) * WIN + 4 * wq));
        const v4f* p1 = (const v4f*)((const float*)p0 + WIN);
        v4f a = __builtin_nontemporal_load(p0);  // row 2*hp, cols 4wq..4wq+3
        v4f b = __builtin_nontemporal_load(p1);  // row 2*hp+1
        float s  = bn_scale[c];
        float bb = bn_bias[c];
        float a0 = fmaxf(fmaf(a.x, s, bb), 0.0f);
        float a1 = fmaxf(fmaf(a.y, s, bb), 0.0f);
        float a2 = fmaxf(fmaf(a.z, s, bb), 0.0f);
        float a3 = fmaxf(fmaf(a.w, s, bb), 0.0f);
        float b0 = fmaxf(fmaf(b.x, s, bb), 0.0f);
        float b1 = fmaxf(fmaf(b.y, s, bb), 0.0f);
        float b2 = fmaxf(fmaf(b.z, s, bb), 0.0f);
        float b3 = fmaxf(fmaf(b.w, s, bb), 0.0f);
        v2f pv;
        pv.x = (a0 + a1 + b0 + b1) * 0.25f;
        pv.y = (a2 + a3 + b2 + b3) * 0.25f;
        *(v2f*)&lds_pool[c * WP + 2 * wq] = pv;
    }

    // ---------- Stage 2: preload B fragments (weights) ----------
    // B 4x16 f32 layout (2 VGPRs): lanes 0-15 hold B[K=0..1][N=lane],
    // lanes 16-31 hold B[K=2..3][N=lane-16]. Weights are reused by all 4096
    // blocks -> keep regular-temporal (cache-resident).
    const int lane = t & 31;
    const int wave = t >> 5;
    const int half = lane >> 4;   // 0: K pair {0,1}; 1: K pair {2,3}
    const int l    = lane & 15;   // N (out-channel) / M (pixel) index
    const int m0   = wave * 16;   // this wave's 16-pixel M tile

    v2f bfrag[8];
    #pragma unroll
    for (int k = 0; k < 8; ++k) {
        int kb = k * 4 + half * 2;
        v2f f;
        f.x = w[(kb + 0) * COUT + l];
        f.y = w[(kb + 1) * COUT + l];
        bfrag[k] = f;
    }

    __syncthreads();

    // ---------- Stage 3: channel GEMM via V_WMMA_F32_16X16X4_F32 ----------
    // A 16x4 f32 layout: lanes 0-15 hold A[M=lane][K=0..1] in 2 VGPRs,
    // lanes 16-31 hold A[M=lane-16][K=2..3].  A[m][k] = pooled[c=k][m0+m].
    v8f acc = {};
    #pragma unroll
    for (int k = 0; k < 8; ++k) {
        int kb = k * 4 + half * 2;
        v2f af;
        af.x = lds_pool[(kb + 0) * WP + m0 + l];
        af.y = lds_pool[(kb + 1) * WP + m0 + l];
        acc = __builtin_amdgcn_wmma_f32_16x16x4_f32(
            /*neg_a=*/false, af, /*neg_b=*/false, bfrag[k],
            /*c_mod=*/(short)0, acc, /*reuse_a=*/false, /*reuse_b=*/false);
    }

    // ---------- Stage 4: D tile -> LDS transpose ----------
    // D layout: VGPR v, lanes 0-15 -> (M=v, N=lane); lanes 16-31 -> (M=v+8, N=lane-16).
    // Consecutive v are contiguous in LDS -> compiler emits ds_store_b128 x2.
    #pragma unroll
    for (int v = 0; v < 8; ++v) {
        int m = m0 + v + half * 8;
        lds_out[l * OUT_PAD + m] = acc[v];
    }

    __syncthreads();

    // ---------- Stage 5: coalesced global store (write-once -> non-temporal) ----------
    #pragma unroll
    for (int i = 0; i < 8; ++i) {
        int idx = i * 256 + t;   // 0..2047
        int d   = idx >> 7;      // 0..15
        int wp  = idx & 127;
        float* dst = out + (((size_t)n * COUT + d) * HP + hp) * WP + wp;
        __builtin_nontemporal_store(lds_out[d * OUT_PAD + wp], dst);
    }
}

extern "C" void kernel_launch(void* const* d_in, const int* in_sizes, int n_in,
                              void* d_out, int out_size, void* d_ws, size_t ws_size,
                              hipStream_t stream) {
    const float* x        = (const float*)d_in[0];
    const float* w        = (const float*)d_in[1];
    const float* bn_scale = (const float*)d_in[2];
    const float* bn_bias  = (const float*)d_in[3];
    float* out = (float*)d_out;

    dim3 grid(32 * 128);   // one block per (n, pooled row)
    dim3 block(256);       // 8 wave32s
    fused_bn_relu_pool_conv1x1<<<grid, block, 0, stream>>>(x, w, bn_scale, bn_bias, out);
}